// IntegratedMoE_34694745817707
// MI455X (gfx1250) — compile-verified
//
#include <hip/hip_runtime.h>
#include <math.h>

typedef __attribute__((ext_vector_type(2))) float v2f;
typedef __attribute__((ext_vector_type(8))) float v8f;

#define E_   4
#define B_   2048
#define Q_   100
#define C_   92
#define H_   16
#define BOX_ 4

// ---------------------------------------------------------------------------
// Kernel 1: probs[b][e] = sigmoid( mean_q logits[e][b][q][0] )
// One wave per (e,b); lanes parallelize over q (stride-C loads pull exactly
// one cacheline per element, the minimum possible for this access pattern).
// ---------------------------------------------------------------------------
__global__ void moe_probs_kernel(const float* __restrict__ logits,
                                 float* __restrict__ probs_out)
{
    const int wave = (blockIdx.x * blockDim.x + threadIdx.x) >> 5;
    const int lane = threadIdx.x & 31;
    if (wave >= E_ * B_) return;
    const int e = wave / B_;
    const int b = wave % B_;
    const float* base = logits + (size_t)(e * B_ + b) * Q_ * C_;

    float s = 0.0f;
    s += base[(size_t)(lane +  0) * C_];
    s += base[(size_t)(lane + 32) * C_];
    s += base[(size_t)(lane + 64) * C_];
    if (lane < 4) s += base[(size_t)(lane + 96) * C_];   // 100 = 3*32 + 4

    // butterfly reduce across the wave32
    for (int off = 16; off > 0; off >>= 1)
        s += __shfl_xor(s, off, 32);

    if (lane == 0) {
        float m = s * (1.0f / (float)Q_);
        probs_out[b * E_ + e] = 1.0f / (1.0f + expf(-m));
    }
}

// ---------------------------------------------------------------------------
// Kernel 2: gating MLP via V_WMMA_F32_16X16X4_F32 (one wave per 16-row tile)
//   layer1: [16x4] @ W1[4x16]            -> 1 WMMA (K=4)
//   layer2: h[16x16] @ W2pad[16x16]      -> 4 accumulating WMMAs (K=4 each)
// then per-row softmax, top-2, renormalize, final_pred.
//
// f32 WMMA operand layout (ISA 7.12.2):
//   A 16x4:  lanes 0-15 hold M=lane, vgpr0/1 = K0/K1; lanes 16-31 = K2/K3
//   B 4x16:  lanes hold N=lane&15,   vgpr0/1 = K{0,1} (lo half) / K{2,3} (hi)
//   C/D:     lane N = lane&15, vgpr r -> M = r + 8*(lane>=16)
//
// W2 B-operands are loaded branch-free (clamped index + select) and hoisted
// to the top of the kernel so the four layer-2 WMMAs can issue back-to-back.
// ---------------------------------------------------------------------------
__global__ void moe_gate_kernel(const float* __restrict__ probs,   // [B,E]
                                const float* __restrict__ W1,      // [E,H]
                                const float* __restrict__ b1,      // [H]
                                const float* __restrict__ W2,      // [H,E]
                                const float* __restrict__ b2,      // [E]
                                float* __restrict__ norm_w_out,    // [B,E]
                                float* __restrict__ final_pred,    // [B]
                                float* __restrict__ top_idx_out,   // [B,2] (as float)
                                float* __restrict__ ws_w,          // [B,2]
                                int*   __restrict__ ws_i)          // [B,2]
{
    __shared__ float hbuf[16 * 16];   // h tile, [m][k] layout for layer-2 A
    __shared__ float zbuf[16 * E_];   // layer-2 output, cols 0..3

    const int lane = threadIdx.x;     // blockDim.x == 32, one wave
    const int b0   = blockIdx.x * 16; // 16 batch rows per tile
    const int m    = lane & 15;
    const int hi   = lane >> 4;       // 0 -> K{0,1}, 1 -> K{2,3}

    // ---- hoisted, branch-free W2 B-operands for all 4 layer-2 K-slices ----
    const int   mc   = m < E_ ? m : (E_ - 1);    // clamped column
    const bool  live = (m < E_);
    v2f w2op[4];
    #pragma unroll
    for (int kk = 0; kk < 4; ++kk) {
        const int k0 = kk * 4 + hi * 2;
        float vx = W2[(k0 + 0) * E_ + mc];       // unconditional load
        float vy = W2[(k0 + 1) * E_ + mc];
        w2op[kk].x = live ? vx : 0.0f;           // v_cndmask pad -> 0
        w2op[kk].y = live ? vy : 0.0f;
    }

    // ---- layer 1: A = probs tile [16x4], B = W1 [4x16] ----
    v2f a, bm;
    a.x  = probs[(b0 + m) * E_ + (hi * 2 + 0)];
    a.y  = probs[(b0 + m) * E_ + (hi * 2 + 1)];
    bm.x = W1[(hi * 2 + 0) * H_ + m];
    bm.y = W1[(hi * 2 + 1) * H_ + m];
    v8f c = {};
    c = __builtin_amdgcn_wmma_f32_16x16x4_f32(false, a, false, bm,
                                              (short)0, c, false, false);

    const float bias1 = b1[m];        // column index n == lane&15
    #pragma unroll
    for (int r = 0; r < 8; ++r) {
        int   row = r + hi * 8;
        float v   = c[r] + bias1;
        hbuf[row * 16 + m] = v > 0.0f ? v : 0.0f;   // ReLU, store as [m][k]
    }
    __syncthreads();

    // ---- layer 2: 4 accumulating K=4 WMMAs over H=16 (back-to-back) ----
    v8f acc = {};
    #pragma unroll
    for (int kk = 0; kk < 4; ++kk) {
        const int k0 = kk * 4 + hi * 2;
        v2f a2;
        a2.x = hbuf[m * 16 + k0 + 0];
        a2.y = hbuf[m * 16 + k0 + 1];
        acc = __builtin_amdgcn_wmma_f32_16x16x4_f32(false, a2, false, w2op[kk],
                                                    (short)0, acc, false, false);
    }

    if (live) {
        const float bias2 = b2[m];
        #pragma unroll
        for (int r = 0; r < 8; ++r) {
            int row = r + hi * 8;
            zbuf[row * E_ + m] = acc[r] + bias2;
        }
    }
    __syncthreads();

    // ---- per-row softmax + top-2 + renormalize (lanes 0-15, one row each) ----
    if (lane < 16) {
        const int b = b0 + lane;
        float w[E_];
        float mx = -3.0e38f;
        #pragma unroll
        for (int e = 0; e < E_; ++e) { w[e] = zbuf[lane * E_ + e]; mx = fmaxf(mx, w[e]); }
        float sum = 0.0f;
        #pragma unroll
        for (int e = 0; e < E_; ++e) { w[e] = expf(w[e] - mx); sum += w[e]; }
        const float inv = 1.0f / sum;
        #pragma unroll
        for (int e = 0; e < E_; ++e) w[e] *= inv;

        // top-2, first occurrence wins on ties (matches lax.top_k)
        int i0 = 0;
        #pragma unroll
        for (int e = 1; e < E_; ++e) if (w[e] > w[i0]) i0 = e;
        int i1 = -1;
        #pragma unroll
        for (int e = 0; e < E_; ++e)
            if (e != i0 && (i1 < 0 || w[e] > w[i1])) i1 = e;

        const float s2   = w[i0] + w[i1];
        const float rinv = 1.0f / (s2 + 1e-8f);
        const float w0   = w[i0] * rinv;
        const float w1   = w[i1] * rinv;

        #pragma unroll
        for (int e = 0; e < E_; ++e) norm_w_out[b * E_ + e] = 0.0f;
        norm_w_out[b * E_ + i0] = w0;
        norm_w_out[b * E_ + i1] = w1;
        final_pred[b] = w0 * probs[b * E_ + i0] + w1 * probs[b * E_ + i1];
        top_idx_out[b * 2 + 0] = (float)i0;
        top_idx_out[b * 2 + 1] = (float)i1;
        ws_w[b * 2 + 0] = w0;  ws_w[b * 2 + 1] = w1;
        ws_i[b * 2 + 0] = i0;  ws_i[b * 2 + 1] = i1;
    }
}

// ---------------------------------------------------------------------------
// Kernel 3: streaming combine (bandwidth-bound phase, ~340 MB total traffic).
//   blockIdx.x in [0,9): logits (Q*C = 9200 floats = 2300 float4 per b)
//   blockIdx.x == 9:     boxes  (Q*4 = 400 floats = 100 float4 per b)
// float4 -> global_load_b128 / global_store_b128.
// One global_prefetch_b8 per 128B cacheline (idx4 % 8 == 0), one b ahead.
// ---------------------------------------------------------------------------
__global__ void moe_combine_kernel(const float* __restrict__ logits,
                                   const float* __restrict__ boxes,
                                   const float* __restrict__ ws_w,
                                   const int*   __restrict__ ws_i,
                                   float* __restrict__ out_logits,
                                   float* __restrict__ out_boxes)
{
    const int   b  = blockIdx.y;
    const int   i0 = ws_i[b * 2 + 0];
    const int   i1 = ws_i[b * 2 + 1];
    const float w0 = ws_w[b * 2 + 0];
    const float w1 = ws_w[b * 2 + 1];

    if (blockIdx.x < 9) {
        const int idx4 = blockIdx.x * blockDim.x + threadIdx.x;
        if (idx4 < (Q_ * C_ / 4)) {
            const size_t off = (size_t)b * (Q_ * C_) + (size_t)idx4 * 4;
            const float* p0 = logits + (size_t)i0 * B_ * Q_ * C_ + off;
            const float* p1 = logits + (size_t)i1 * B_ * Q_ * C_ + off;
            if ((idx4 & 7) == 0 && (b + 1) < B_) {   // 1 prefetch per cacheline
                __builtin_prefetch(p0 + (Q_ * C_), 0, 1);
                __builtin_prefetch(p1 + (Q_ * C_), 0, 1);
            }
            const float4 x0 = *(const float4*)p0;
            const float4 x1 = *(const float4*)p1;
            float4 r;
            r.x = w0 * x0.x + w1 * x1.x;
            r.y = w0 * x0.y + w1 * x1.y;
            r.z = w0 * x0.z + w1 * x1.z;
            r.w = w0 * x0.w + w1 * x1.w;
            *(float4*)(out_logits + off) = r;
        }
    } else {
        const int idx4 = threadIdx.x;
        if (idx4 < (Q_ * BOX_ / 4)) {
            const size_t off = (size_t)b * (Q_ * BOX_) + (size_t)idx4 * 4;
            const float4 x0 = *(const float4*)(boxes + (size_t)i0 * B_ * Q_ * BOX_ + off);
            const float4 x1 = *(const float4*)(boxes + (size_t)i1 * B_ * Q_ * BOX_ + off);
            float4 r;
            r.x = w0 * x0.x + w1 * x1.x;
            r.y = w0 * x0.y + w1 * x1.y;
            r.z = w0 * x0.z + w1 * x1.z;
            r.w = w0 * x0.w + w1 * x1.w;
            *(float4*)(out_boxes + off) = r;
        }
    }
}

// ---------------------------------------------------------------------------
extern "C" void kernel_launch(void* const* d_in, const int* in_sizes, int n_in,
                              void* d_out, int out_size, void* d_ws, size_t ws_size,
                              hipStream_t stream)
{
    const float* logits = (const float*)d_in[0];   // [E,B,Q,C]
    const float* boxes  = (const float*)d_in[1];   // [E,B,Q,4]
    const float* W1     = (const float*)d_in[2];   // [E,H]
    const float* b1     = (const float*)d_in[3];   // [H]
    const float* W2     = (const float*)d_in[4];   // [H,E]
    const float* b2     = (const float*)d_in[5];   // [E]

    float* out        = (float*)d_out;
    float* out_logits = out;                                   // B*Q*C   = 18,841,600
    float* out_boxes  = out_logits + (size_t)B_ * Q_ * C_;     // B*Q*4   =    819,200
    float* out_pred   = out_boxes  + (size_t)B_ * Q_ * BOX_;   // B       =      2,048
    float* out_normw  = out_pred   + B_;                       // B*E     =      8,192
    float* out_probs  = out_normw  + B_ * E_;                  // B*E     =      8,192
    float* out_topidx = out_probs  + B_ * E_;                  // B*2     =      4,096

    float* ws_w = (float*)d_ws;                                // [B,2]
    int*   ws_i = (int*)((char*)d_ws + (size_t)B_ * 2 * sizeof(float));

    // 1) probs: one wave per (e,b) -> 8192 waves, 8 waves per 256-thread block
    moe_probs_kernel<<<dim3((E_ * B_) / 8), 256, 0, stream>>>(logits, out_probs);

    // 2) gating MLP via f32 WMMA: one wave per 16 batch rows
    moe_gate_kernel<<<dim3(B_ / 16), 32, 0, stream>>>(out_probs, W1, b1, W2, b2,
                                                      out_normw, out_pred, out_topidx,
                                                      ws_w, ws_i);

    // 3) streaming combine: grid (10, B); x<9 -> logits, x==9 -> boxes
    moe_combine_kernel<<<dim3(10, B_), 256, 0, stream>>>(logits, boxes, ws_w, ws_i,
                                                         out_logits, out_boxes);
}